// GraphSAGE_31224412242363
// MI455X (gfx1250) — compile-verified
//
#include <hip/hip_runtime.h>
#include <hip/hip_bf16.h>

typedef __attribute__((ext_vector_type(2))) float v2f;
typedef __attribute__((ext_vector_type(8))) float v8f;

#define NNODES 50000
#define DFEAT  128
#define LDSPITCH 132   // 128 + 4 padding floats -> bank-conflict-free column reads

// ---------------- utility kernels ----------------

__global__ void zero_f32(float* __restrict__ p, int n) {
    int i = blockIdx.x * blockDim.x + threadIdx.x;
    if (i < n) p[i] = 0.0f;
}

__global__ void count_deg(const int* __restrict__ dst, float* __restrict__ deg, int E) {
    int i = blockIdx.x * blockDim.x + threadIdx.x;
    if (i < E) atomicAdd(&deg[dst[i]], 1.0f);
}

__global__ void recip_deg(float* __restrict__ deg, int n) {
    int i = blockIdx.x * blockDim.x + threadIdx.x;
    if (i < n) deg[i] = 1.0f / fmaxf(deg[i], 1.0f);
}

// One wave (32 lanes) per edge; lane handles 4 consecutive features.
// hn[dst] += h[src] * invdeg[dst]   (mean aggregation, pre-scaled scatter)
__global__ void scatter_mean(const float* __restrict__ h,
                             const int* __restrict__ src,
                             const int* __restrict__ dst,
                             const float* __restrict__ invdeg,
                             float* __restrict__ hn, int E) {
    int tid = blockIdx.x * blockDim.x + threadIdx.x;
    int e = tid >> 5;
    if (e >= E) return;
    int c4 = (tid & 31) << 2;
    int s = src[e];
    int d = dst[e];
    float w = invdeg[d];
    const float4 v = *(const float4*)(h + (size_t)s * DFEAT + c4);
    float* o = hn + (size_t)d * DFEAT + c4;
    atomicAdd(o + 0, v.x * w);
    atomicAdd(o + 1, v.y * w);
    atomicAdd(o + 2, v.z * w);
    atomicAdd(o + 3, v.w * w);
}

// ---------------- fused SAGE GEMM via V_WMMA_F32_16X16X4_F32 ----------------
// out[16 rows x 128 cols] = relu?( 2*h@Ws + hn@Wn + 2*b )
// Block = 256 threads = 8 waves; wave w computes 16x16 tile at cols [16w,16w+16).
__global__ __launch_bounds__(256)
void sage_gemm(const float* __restrict__ hs,   // [N,128] self features
               const float* __restrict__ hn,   // [N,128] aggregated neighbor features
               const float* __restrict__ Ws,   // [128,128]
               const float* __restrict__ Wn,   // [128,128]
               const float* __restrict__ bias, // [128]
               float* __restrict__ out,        // [N,128]
               int relu) {
    __shared__ float lds_h[16 * LDSPITCH];
    __shared__ float lds_n[16 * LDSPITCH];

    const int row0 = blockIdx.x * 16;
    const int t = threadIdx.x;

    // Cooperative stage of the 16x128 A panels (h and hn) into LDS.
    // 512 float4 per panel, 2 per thread.
    #pragma unroll
    for (int i = 0; i < 2; ++i) {
        int idx = t + i * 256;
        int r   = idx >> 5;          // row 0..15
        int c4  = (idx & 31) << 2;   // col 0..124 step 4
        const float4 vh = *(const float4*)(hs + (size_t)(row0 + r) * DFEAT + c4);
        const float4 vn = *(const float4*)(hn + (size_t)(row0 + r) * DFEAT + c4);
        *(float4*)(&lds_h[r * LDSPITCH + c4]) = vh;
        *(float4*)(&lds_n[r * LDSPITCH + c4]) = vn;
    }
    __syncthreads();

    const int wave = t >> 5;
    const int lane = t & 31;
    const int col0 = wave * 16;
    const int mn   = lane & 15;            // A row / B col within tile
    const int koff = (lane >> 4) << 1;     // lanes 0-15 -> K{0,1}; lanes 16-31 -> K{2,3}

    v8f acc = {};

    // 2 * h @ Ws   (self term appears once per edge type -> factor 2 folded into A)
    #pragma unroll
    for (int k0 = 0; k0 < DFEAT; k0 += 4) {
        const int k = k0 + koff;
        v2f a, b;
        a.x = 2.0f * lds_h[mn * LDSPITCH + k];
        a.y = 2.0f * lds_h[mn * LDSPITCH + k + 1];
        b.x = Ws[(size_t)k * DFEAT + col0 + mn];
        b.y = Ws[(size_t)(k + 1) * DFEAT + col0 + mn];
        acc = __builtin_amdgcn_wmma_f32_16x16x4_f32(false, a, false, b,
                                                    (short)0, acc, false, false);
    }
    // hn @ Wn
    #pragma unroll
    for (int k0 = 0; k0 < DFEAT; k0 += 4) {
        const int k = k0 + koff;
        v2f a, b;
        a.x = lds_n[mn * LDSPITCH + k];
        a.y = lds_n[mn * LDSPITCH + k + 1];
        b.x = Wn[(size_t)k * DFEAT + col0 + mn];
        b.y = Wn[(size_t)(k + 1) * DFEAT + col0 + mn];
        acc = __builtin_amdgcn_wmma_f32_16x16x4_f32(false, a, false, b,
                                                    (short)0, acc, false, false);
    }

    // Epilogue. C/D layout: VGPR v, lanes 0-15 -> M=v, lanes 16-31 -> M=8+v; N=lane&15.
    const int mbase = (lane < 16) ? 0 : 8;
    const float bb = 2.0f * bias[col0 + mn];
    #pragma unroll
    for (int v = 0; v < 8; ++v) {
        float val = acc[v] + bb;
        if (relu) val = fmaxf(val, 0.0f);
        out[(size_t)(row0 + mbase + v) * DFEAT + col0 + mn] = val;
    }
}

// ---------------- launch ----------------

extern "C" void kernel_launch(void* const* d_in, const int* in_sizes, int n_in,
                              void* d_out, int out_size, void* d_ws, size_t ws_size,
                              hipStream_t stream) {
    const float* h    = (const float*)d_in[0];
    const int* src0   = (const int*)d_in[1];
    const int* dst0   = (const int*)d_in[2];
    const int* src1   = (const int*)d_in[3];
    const int* dst1   = (const int*)d_in[4];
    const float* Ws1  = (const float*)d_in[5];
    const float* Wn1  = (const float*)d_in[6];
    const float* b1   = (const float*)d_in[7];
    const float* Ws2  = (const float*)d_in[8];
    const float* Wn2  = (const float*)d_in[9];
    const float* b2   = (const float*)d_in[10];
    float* out = (float*)d_out;

    const int E = in_sizes[1];
    const int N = NNODES;

    // workspace layout: deg0[N] | deg1[N] | hn[N*D] | h1[N*D]
    float* deg0 = (float*)d_ws;
    float* deg1 = deg0 + N;
    float* hn   = deg1 + N;
    float* h1   = hn + (size_t)N * DFEAT;

    const int ZN1 = 2 * N + N * DFEAT;     // deg0|deg1|hn contiguous
    const int ND  = N * DFEAT;
    const int scatterThreads = E * 32;
    const int rowBlocks = N / 16;          // 3125

    // ---- degree pass (shared by both layers) ----
    zero_f32<<<(ZN1 + 255) / 256, 256, 0, stream>>>(deg0, ZN1);
    count_deg<<<(E + 255) / 256, 256, 0, stream>>>(dst0, deg0, E);
    count_deg<<<(E + 255) / 256, 256, 0, stream>>>(dst1, deg1, E);
    recip_deg<<<(2 * N + 255) / 256, 256, 0, stream>>>(deg0, 2 * N);

    // ---- layer 1 ----
    scatter_mean<<<(scatterThreads + 255) / 256, 256, 0, stream>>>(h, src0, dst0, deg0, hn, E);
    scatter_mean<<<(scatterThreads + 255) / 256, 256, 0, stream>>>(h, src1, dst1, deg1, hn, E);
    sage_gemm<<<rowBlocks, 256, 0, stream>>>(h, hn, Ws1, Wn1, b1, h1, 1);

    // ---- layer 2 ----
    zero_f32<<<(ND + 255) / 256, 256, 0, stream>>>(hn, ND);
    scatter_mean<<<(scatterThreads + 255) / 256, 256, 0, stream>>>(h1, src0, dst0, deg0, hn, E);
    scatter_mean<<<(scatterThreads + 255) / 256, 256, 0, stream>>>(h1, src1, dst1, deg1, hn, E);
    sage_gemm<<<rowBlocks, 256, 0, stream>>>(h1, hn, Ws2, Wn2, b2, out, 0);
}